// LogicDenseCuda_5196910428686
// MI455X (gfx1250) — compile-verified
//
#include <hip/hip_runtime.h>

typedef __attribute__((ext_vector_type(2))) float v2f;
typedef __attribute__((ext_vector_type(8))) float v8f;

// op_k(a,b) = C[k][0] + C[k][1]*a + C[k][2]*b + C[k][3]*a*b
__constant__ float OPC[16][4] = {
    {0.f,  0.f,  0.f,  0.f},
    {0.f,  0.f,  0.f,  1.f},
    {0.f,  1.f,  0.f, -1.f},
    {0.f,  1.f,  0.f,  0.f},
    {0.f,  0.f,  1.f, -1.f},
    {0.f,  0.f,  1.f,  0.f},
    {0.f,  1.f,  1.f, -2.f},
    {0.f,  1.f,  1.f, -1.f},
    {1.f, -1.f, -1.f,  1.f},
    {1.f, -1.f, -1.f,  2.f},
    {1.f,  0.f, -1.f,  0.f},
    {1.f,  0.f, -1.f,  1.f},
    {1.f, -1.f,  0.f,  0.f},
    {1.f, -1.f,  0.f,  1.f},
    {1.f,  0.f,  0.f, -1.f},
    {1.f,  0.f,  0.f,  0.f},
};

// OP_COEFFS rows packed as 4 signed int8 per gate: byte j = C[g][j].
// Extract column n via (pk << (3-n)*8) >> 24 (arithmetic shift sign-extends).
#define PK8(c0, c1, c2, c3)                                            \
  (int)(((unsigned)(unsigned char)(char)(c0)) |                        \
        (((unsigned)(unsigned char)(char)(c1)) << 8) |                 \
        (((unsigned)(unsigned char)(char)(c2)) << 16) |                \
        (((unsigned)(unsigned char)(char)(c3)) << 24))

__device__ constexpr int PKC[16] = {
    PK8(0, 0, 0, 0),   PK8(0, 0, 0, 1),   PK8(0, 1, 0, -1),  PK8(0, 1, 0, 0),
    PK8(0, 0, 1, -1),  PK8(0, 0, 1, 0),   PK8(0, 1, 1, -2),  PK8(0, 1, 1, -1),
    PK8(1, -1, -1, 1), PK8(1, -1, -1, 2), PK8(1, 0, -1, 0),  PK8(1, 0, -1, 1),
    PK8(1, -1, 0, 0),  PK8(1, -1, 0, 1),  PK8(1, 0, 0, -1),  PK8(1, 0, 0, 0),
};

// ---------------------------------------------------------------------------
// Stage 1: q[neuron][0..3] = softmax(weight[neuron][0..15]) @ OPC
// One wave per tile of 16 neurons. q tile = 4 accumulating
// V_WMMA_F32_16X16X4_F32: D(16x16) += A(16x4) * B(4x16), K=16 gates split
// into 4 chunks of 4. Exact f32 arithmetic, matches the f32 reference.
// All operand construction uses compile-time indices + single selects so no
// cndmask chains / exec-masked loads appear before the WMMAs (EXEC all-1s).
// ---------------------------------------------------------------------------
__global__ __launch_bounds__(32) void qprep_wmma_kernel(
    const float* __restrict__ weight, float* __restrict__ qout) {
  const int lane = threadIdx.x & 31;
  const int m    = lane & 15;        // neuron-in-tile (A rows); also N col
  const bool hh  = (lane >> 4) != 0; // lane-half selector for K striping
  const int tile = blockIdx.x;
  const int neuron = tile * 16 + m;

  // Each lane owns one neuron's 16 logits (64B contiguous row).
  const float* wrow = weight + (size_t)neuron * 16;
  float w[16];
#pragma unroll
  for (int g = 0; g < 16; ++g) w[g] = wrow[g];

  float mx = w[0];
#pragma unroll
  for (int g = 1; g < 16; ++g) mx = fmaxf(mx, w[g]);

  float e[16];
  float s = 0.f;
#pragma unroll
  for (int g = 0; g < 16; ++g) {
    e[g] = expf(w[g] - mx);
    s += e[g];
  }
  const float inv = 1.0f / s;

  const int n      = m;            // B/D column index; only n<4 columns real
  const bool ncol  = (n < 4);
  const int  shamt = (3 - (n & 3)) * 8;

  v8f acc = {};

#if __has_builtin(__builtin_amdgcn_wmma_f32_16x16x4_f32)
#pragma unroll
  for (int c = 0; c < 4; ++c) {
    // 32-bit A 16x4 layout: lanes 0-15 hold K={0,1}, lanes 16-31 K={2,3}.
    // Both candidates are compile-time elements -> one cndmask each.
    const float a0 = hh ? e[4 * c + 2] : e[4 * c + 0];
    const float a1 = hh ? e[4 * c + 3] : e[4 * c + 1];
    v2f Av;
    Av.x = a0 * inv;
    Av.y = a1 * inv;
    // 32-bit B 4x16: same K striping; coefficient from packed-int8 immediates.
    const int p0 = hh ? PKC[4 * c + 2] : PKC[4 * c + 0];
    const int p1 = hh ? PKC[4 * c + 3] : PKC[4 * c + 1];
    const float b0 = (float)((p0 << shamt) >> 24);
    const float b1 = (float)((p1 << shamt) >> 24);
    v2f Bv;
    Bv.x = ncol ? b0 : 0.0f;
    Bv.y = ncol ? b1 : 0.0f;
    acc = __builtin_amdgcn_wmma_f32_16x16x4_f32(
        /*neg_a=*/false, Av, /*neg_b=*/false, Bv,
        /*c_mod=*/(short)0, acc, /*reuse_a=*/false, /*reuse_b=*/false);
  }
  // f32 C/D 16x16 layout: VGPR v -> M = v + 8*h, N = lane&15
  if (ncol) {
    const int mbase = tile * 16 + (hh ? 8 : 0);
#pragma unroll
    for (int v = 0; v < 8; ++v) {
      qout[(size_t)(mbase + v) * 4 + n] = acc[v];
    }
  }
#else
  // Fallback (no WMMA builtin): lane-half h==0 writes its own neuron's q.
  if (!hh) {
#pragma unroll
    for (int j = 0; j < 4; ++j) {
      float acc_s = 0.f;
#pragma unroll
      for (int g = 0; g < 16; ++g) acc_s = fmaf(e[g] * inv, OPC[g][j], acc_s);
      qout[(size_t)neuron * 4 + j] = acc_s;
    }
  }
#endif
}

// Scalar tail for out_dim not divisible by 16 (not hit for the given shapes).
__global__ void qprep_tail_kernel(const float* __restrict__ weight,
                                  float* __restrict__ qout, int start,
                                  int out_dim) {
  int neuron = start + blockIdx.x * blockDim.x + threadIdx.x;
  if (neuron >= out_dim) return;
  const float* wrow = weight + (size_t)neuron * 16;
  float w[16];
#pragma unroll
  for (int g = 0; g < 16; ++g) w[g] = wrow[g];
  float mx = w[0];
#pragma unroll
  for (int g = 1; g < 16; ++g) mx = fmaxf(mx, w[g]);
  float e[16], s = 0.f;
#pragma unroll
  for (int g = 0; g < 16; ++g) {
    e[g] = expf(w[g] - mx);
    s += e[g];
  }
  float inv = 1.0f / s;
#pragma unroll
  for (int j = 0; j < 4; ++j) {
    float acc = 0.f;
#pragma unroll
    for (int g = 0; g < 16; ++g) acc = fmaf(e[g] * inv, OPC[g][j], acc);
    qout[(size_t)neuron * 4 + j] = acc;
  }
}

// ---------------------------------------------------------------------------
// Stage 2: out[i][j] = q[j].x + q[j].y*a + q[j].z*b + q[j].w*a*b
// Thread owns column j, loops IT batch rows: q/indices load once, x-row
// gathers hit L2 (64 MB x fits in 192 MB L2), output streamed with NT
// stores so the 268 MB output doesn't evict the L2-resident x.
// ---------------------------------------------------------------------------
template <int IT>
__global__ __launch_bounds__(256) void logic_main_kernel(
    const float* __restrict__ x, const float* __restrict__ q,
    const int* __restrict__ indices, float* __restrict__ out, int in_dim,
    int out_dim, int i_off) {
  const int j = blockIdx.x * blockDim.x + threadIdx.x;
  if (j >= out_dim) return;

  const float4 qj = ((const float4*)q)[j];
  const int ia = indices[j];
  const int ib = indices[out_dim + j];

  const int i0 = i_off + blockIdx.y * IT;
  const float* xrow = x + (size_t)i0 * in_dim;
  float* optr = out + (size_t)i0 * out_dim + j;

#pragma unroll
  for (int t = 0; t < IT; ++t) {
    const float a = xrow[ia];
    const float b = xrow[ib];
    const float r = fmaf(a * b, qj.w, fmaf(b, qj.z, fmaf(a, qj.y, qj.x)));
    __builtin_nontemporal_store(r, optr);
    xrow += in_dim;
    optr += out_dim;
  }
}

extern "C" void kernel_launch(void* const* d_in, const int* in_sizes, int n_in,
                              void* d_out, int out_size, void* d_ws,
                              size_t ws_size, hipStream_t stream) {
  const float* x       = (const float*)d_in[0];
  const float* weight  = (const float*)d_in[1];
  const int*   indices = (const int*)d_in[2];
  float*       out     = (float*)d_out;

  const int out_dim = in_sizes[1] / 16;          // weight is (out_dim, 16)
  const int batch   = out_size / out_dim;        // out is (batch, out_dim)
  const int in_dim  = in_sizes[0] / batch;       // x is (batch, in_dim)

  float* q = (float*)d_ws;                       // out_dim * 4 floats

  // Stage 1: WMMA softmax-matmul for q.
  const int tiles = out_dim / 16;
  if (tiles > 0) {
    qprep_wmma_kernel<<<tiles, 32, 0, stream>>>(weight, q);
  }
  const int qrem = out_dim % 16;
  if (qrem > 0) {
    qprep_tail_kernel<<<1, 64, 0, stream>>>(weight, q, tiles * 16, out_dim);
  }

  // Stage 2: streaming gather/affine, NT stores.
  constexpr int IT = 8;
  const dim3 block(256);
  const int jblocks = (out_dim + 255) / 256;
  const int ifull = batch / IT;
  if (ifull > 0) {
    dim3 grid(jblocks, ifull);
    logic_main_kernel<IT><<<grid, block, 0, stream>>>(x, q, indices, out,
                                                      in_dim, out_dim, 0);
  }
  const int itail = batch % IT;
  if (itail > 0) {
    dim3 grid(jblocks, itail);
    logic_main_kernel<1><<<grid, block, 0, stream>>>(
        x, q, indices, out, in_dim, out_dim, ifull * IT);
  }
}